// MAF_3461743640885
// MI455X (gfx1250) — compile-verified
//
#include <hip/hip_runtime.h>
#include <math.h>

#define DIMV   64
#define HIDV   64
#define LVAL   63
#define BATCHV 16384

typedef float v2f __attribute__((ext_vector_type(2)));
typedef float v8f __attribute__((ext_vector_type(8)));
typedef unsigned int u32x4 __attribute__((ext_vector_type(4)));
typedef int i32x4 __attribute__((ext_vector_type(4)));
typedef int i32x8 __attribute__((ext_vector_type(8)));

// hardware transcendental tanh on gfx1250 if the builtin exists
#if __has_builtin(__builtin_amdgcn_tanhf)
#define TANHF(x) __builtin_amdgcn_tanhf(x)
#else
#define TANHF(x) tanhf(x)
#endif

#define TSTR  68   // transposed LDS row stride (floats): bank = (4m+kb)%64 -> conflict free
#define XSTR  68   // staging tile row stride (272B = 17*16B, float4-aligned)

#define W1T_ELEMS (63 * 64 * 64)
#define W2T_ELEMS (63 * 64 * 64)
#define W3T_ELEMS (63 * 16 * 64)

// ---------------------------------------------------------------------------
// TDM: 1-D contiguous global -> LDS copy with hardware padding (4 dwords pad
// after every 64 dwords => LDS stride 68 floats). D# per ISA 08 sec 8.3/8.4.
// ---------------------------------------------------------------------------
__device__ __forceinline__ void tdm_load_1d_padded(unsigned lds_byte_addr,
                                                   const float* gsrc,
                                                   unsigned nelem /* dwords, <= 65535 */)
{
    unsigned long long ga = (unsigned long long)(uintptr_t)gsrc;
    u32x4 g0;
    g0[0] = 1u;                                   // count=1 (valid user descriptor)
    g0[1] = lds_byte_addr;                        // lds_addr  [63:32]
    g0[2] = (unsigned)(ga & 0xFFFFFFFFu);         // global_addr[95:64]
    g0[3] = (unsigned)((ga >> 32) & 0x1FFFFFFu)   // global_addr[120:96]
          | (2u << 30);                           // type=2 ("image")
    i32x8 g1;
    g1[0] = (int)((2u << 16)      // data_size = 4 bytes
                | (1u << 20)      // pad_enable
                | (5u << 22)      // pad_interval: 2^(5+1)=64 dwords
                | (3u << 25));    // pad_amount: 3+1 = 4 dwords
    g1[1] = (int)((nelem & 0xFFFFu) << 16);       // tensor_dim0[15:0]
    g1[2] = (int)((nelem >> 16) & 0xFFFFu);       // tensor_dim0[31:16]; tensor_dim1=0
    g1[3] = (int)((nelem & 0xFFFFu) << 16);       // tile_dim0 = nelem (1-D tile)
    g1[4] = 0;                                    // tile_dim1=0, tile_dim2=0 (unused)
    g1[5] = (int)nelem;                           // tensor_dim0_stride[31:0]
    g1[6] = 0;
    g1[7] = 0;
    i32x4 z4 = (i32x4){0, 0, 0, 0};
#if __has_include(<hip/amd_detail/amd_gfx1250_TDM.h>)
    i32x8 z8 = (i32x8){0, 0, 0, 0, 0, 0, 0, 0};   // clang-23 / therock headers: 6-arg form
    __builtin_amdgcn_tensor_load_to_lds(g0, g1, z4, z4, z8, 0);
#else
    __builtin_amdgcn_tensor_load_to_lds(g0, g1, z4, z4, 0);  // ROCm 7.2: 5-arg form
#endif
}

// ---------------------------------------------------------------------------
// Prep: bake mask / K-pad / N-pad and transpose weights into d_ws so the TDM
// can stream layer slabs that land in LDS already in B-fragment-friendly
// [n][k] order:  w1t[l][n][k] = mask(k<=l) * W1[l][k][n]   (k=63 pad -> 0)
//                w2t[l][n][k] = W2[l][k][n]
//                w3t[l][n][k] = (n<2) ? W3[l][k][n] : 0    (n padded to 16)
// ---------------------------------------------------------------------------
__global__ __launch_bounds__(256) void maf_prep_kernel(
    const float* __restrict__ W1g, const float* __restrict__ W2g,
    const float* __restrict__ W3g,
    float* __restrict__ w1t, float* __restrict__ w2t, float* __restrict__ w3t)
{
    int i = blockIdx.x * 256 + threadIdx.x;
    if (i < W1T_ELEMS) {
        int k = i & 63, n = (i >> 6) & 63, l = i >> 12;
        w1t[i] = (k <= l) ? W1g[((l * 63 + k) << 6) + n] : 0.0f;
        w2t[i] = W2g[((l * 64 + k) << 6) + n];
    }
    if (i < W3T_ELEMS) {
        int k = i & 63, n = (i >> 6) & 15, l = i >> 10;
        w3t[i] = (n < 2) ? W3g[((l * 64 + k) << 1) + n] : 0.0f;
    }
}

// ---------------------------------------------------------------------------
// Main kernel: TDM double-buffered weights + f32 WMMA + hw tanh
// ---------------------------------------------------------------------------
__global__ __launch_bounds__(128) void maf_wmma_tdm_kernel(
    const float* __restrict__ xg,
    const float* __restrict__ ip,
    const float* __restrict__ w1t,
    const float* __restrict__ b1g,
    const float* __restrict__ w2t,
    const float* __restrict__ b2g,
    const float* __restrict__ w3t,
    const float* __restrict__ b3g,
    float* __restrict__ outg)
{
    __shared__ float W1s[2][64 * TSTR];   // [n][k] transposed, padded
    __shared__ float W2s[2][64 * TSTR];
    __shared__ float W3s[2][16 * TSTR];
    __shared__ float xs[4][16 * XSTR];
    __shared__ float hs[4][16 * XSTR];
    __shared__ float zs[4][16 * XSTR];
    __shared__ float os[4][16 * 2];

    const int tid  = threadIdx.x;
    const int wid  = tid >> 5;
    const int lane = tid & 31;
    const int m    = lane & 15;
    const int hi   = lane >> 4;

    const int rw = blockIdx.x * 64 + wid * 16;

    float* xw = xs[wid];
    float* hw = hs[wid];
    float* zw = zs[wid];
    float* ow = os[wid];

    // prologue: wave 0 kicks off TDM for layer 0 into buffer 0
    if (wid == 0) {
        tdm_load_1d_padded((unsigned)(uintptr_t)&W1s[0][0], w1t, 64 * 64);
        tdm_load_1d_padded((unsigned)(uintptr_t)&W2s[0][0], w2t, 64 * 64);
        tdm_load_1d_padded((unsigned)(uintptr_t)&W3s[0][0], w3t, 16 * 64);
    }

    // load this wave's 16x64 x tile (coalesced float4)
    for (int i = lane; i < 16 * 16; i += 32) {
        int r = i >> 4, c4 = i & 15;
        float4 v = *(const float4*)&xg[(size_t)(rw + r) * DIMV + c4 * 4];
        *(float4*)&xw[r * XSTR + c4 * 4] = v;
    }

    const float mu0 = ip[0], a0 = ip[1];
    float ld = -a0;
    __syncthreads();  // x tiles staged (and spaces out prologue)

    // persistent X A-fragments: lane holds row m, K = 4s + 2*hi + {0,1}
    v2f xa[16];
#pragma unroll
    for (int s = 0; s < 16; ++s)
        xa[s] = *(const v2f*)&xw[m * XSTR + 4 * s + 2 * hi];

    if (lane < 16)
        zw[lane * XSTR + 63] = (xw[lane * XSTR + 0] - mu0) * expf(-a0);

    for (int l = 0; l < LVAL; ++l) {
        const int cur = l & 1;
        if (wid == 0) __builtin_amdgcn_s_wait_tensorcnt(0);  // buf[cur] complete
        __syncthreads();  // all waves: buf[cur] ready, buf[cur^1] free

        // overlap next layer's DMA with this layer's compute
        if (wid == 0 && l + 1 < LVAL) {
            tdm_load_1d_padded((unsigned)(uintptr_t)&W1s[cur ^ 1][0], w1t + (size_t)(l + 1) * 4096, 64 * 64);
            tdm_load_1d_padded((unsigned)(uintptr_t)&W2s[cur ^ 1][0], w2t + (size_t)(l + 1) * 4096, 64 * 64);
            tdm_load_1d_padded((unsigned)(uintptr_t)&W3s[cur ^ 1][0], w3t + (size_t)(l + 1) * 1024, 16 * 64);
        }

        const float* W1c = W1s[cur];
        const float* W2c = W2s[cur];
        const float* W3c = W3s[cur];

        // ---- layer 1: h1 = tanh(X @ W1_masked + b1)
        v8f acc[4];
#pragma unroll
        for (int t = 0; t < 4; ++t) acc[t] = (v8f){0, 0, 0, 0, 0, 0, 0, 0};
#pragma unroll
        for (int s = 0; s < 16; ++s) {
            v2f a = xa[s];
            const int kb = 4 * s + 2 * hi;
#pragma unroll
            for (int t = 0; t < 4; ++t) {
                v2f b = *(const v2f*)&W1c[(16 * t + m) * TSTR + kb];  // single ds_load_b64
                acc[t] = __builtin_amdgcn_wmma_f32_16x16x4_f32(
                    false, a, false, b, (short)0, acc[t], false, false);
            }
        }
#pragma unroll
        for (int t = 0; t < 4; ++t) {
            float bias = b1g[l * 64 + 16 * t + m];   // lane-broadcast global read
#pragma unroll
            for (int v = 0; v < 8; ++v)
                hw[(v + 8 * hi) * XSTR + 16 * t + m] = TANHF(acc[t][v] + bias);
        }
        v2f ha[16];
#pragma unroll
        for (int s = 0; s < 16; ++s)
            ha[s] = *(const v2f*)&hw[m * XSTR + 4 * s + 2 * hi];

        // ---- layer 2: h2 = tanh(h1 @ W2 + b2)
#pragma unroll
        for (int t = 0; t < 4; ++t) acc[t] = (v8f){0, 0, 0, 0, 0, 0, 0, 0};
#pragma unroll
        for (int s = 0; s < 16; ++s) {
            v2f a = ha[s];
            const int kb = 4 * s + 2 * hi;
#pragma unroll
            for (int t = 0; t < 4; ++t) {
                v2f b = *(const v2f*)&W2c[(16 * t + m) * TSTR + kb];
                acc[t] = __builtin_amdgcn_wmma_f32_16x16x4_f32(
                    false, a, false, b, (short)0, acc[t], false, false);
            }
        }
#pragma unroll
        for (int t = 0; t < 4; ++t) {
            float bias = b2g[l * 64 + 16 * t + m];
#pragma unroll
            for (int v = 0; v < 8; ++v)
                hw[(v + 8 * hi) * XSTR + 16 * t + m] = TANHF(acc[t][v] + bias);
        }
        v2f ha2[16];
#pragma unroll
        for (int s = 0; s < 16; ++s)
            ha2[s] = *(const v2f*)&hw[m * XSTR + 4 * s + 2 * hi];

        // ---- layer 3: out = h2 @ W3 (N padded 2->16)
        v8f acc3 = (v8f){0, 0, 0, 0, 0, 0, 0, 0};
#pragma unroll
        for (int s = 0; s < 16; ++s) {
            const int kb = 4 * s + 2 * hi;
            v2f b = *(const v2f*)&W3c[m * TSTR + kb];
            acc3 = __builtin_amdgcn_wmma_f32_16x16x4_f32(
                false, ha2[s], false, b, (short)0, acc3, false, false);
        }
        if (m == 0) {
#pragma unroll
            for (int v = 0; v < 8; ++v) ow[(v + 8 * hi) * 2 + 0] = acc3[v];
        }
        if (m == 1) {
#pragma unroll
            for (int v = 0; v < 8; ++v) ow[(v + 8 * hi) * 2 + 1] = acc3[v];
        }
        __syncthreads();  // ow visible (also spaces weight-buffer reuse)

        if (lane < 16) {
            float mu = ow[lane * 2 + 0] + b3g[2 * l + 0];
            float al = ow[lane * 2 + 1] + b3g[2 * l + 1];
            float zv = (xw[lane * XSTR + (l + 1)] - mu) * expf(-al);
            zw[lane * XSTR + (62 - l)] = zv;
            ld -= al;
        }
    }

    // coalesced writeback of reversed z + log_det
    for (int i = lane; i < 16 * 16; i += 32) {
        int r = i >> 4, c4 = i & 15;
        float4 v = *(const float4*)&zw[r * XSTR + c4 * 4];
        *(float4*)&outg[(size_t)(rw + r) * DIMV + c4 * 4] = v;
    }
    if (lane < 16)
        outg[(size_t)BATCHV * DIMV + rw + lane] = ld;
}

// ---------------------------------------------------------------------------
// Fallback (no workspace): round-1 kernel, manual LDS staging, still WMMA.
// ---------------------------------------------------------------------------
#define WSTR  80
#define W3STR 20

__global__ __launch_bounds__(128) void maf_wmma_fallback_kernel(
    const float* __restrict__ xg, const float* __restrict__ ip,
    const float* __restrict__ W1g, const float* __restrict__ b1g,
    const float* __restrict__ W2g, const float* __restrict__ b2g,
    const float* __restrict__ W3g, const float* __restrict__ b3g,
    float* __restrict__ outg)
{
    __shared__ float W1s[64 * WSTR];
    __shared__ float W2s[64 * WSTR];
    __shared__ float W3s[64 * W3STR];
    __shared__ float b1s[64], b2s[64], b3s[2];
    __shared__ float xs[4][16 * XSTR];
    __shared__ float hs[4][16 * XSTR];
    __shared__ float zs[4][16 * XSTR];
    __shared__ float os[4][16 * 2];

    const int tid = threadIdx.x, wid = tid >> 5, lane = tid & 31;
    const int m = lane & 15, hi = lane >> 4;
    const int rw = blockIdx.x * 64 + wid * 16;
    float *xw = xs[wid], *hw = hs[wid], *zw = zs[wid], *ow = os[wid];

    for (int i = tid; i < 64 * W3STR; i += 128) W3s[i] = 0.0f;
    for (int i = lane; i < 16 * 16; i += 32) {
        int r = i >> 4, c4 = i & 15;
        float4 v = *(const float4*)&xg[(size_t)(rw + r) * DIMV + c4 * 4];
        *(float4*)&xw[r * XSTR + c4 * 4] = v;
    }
    __syncthreads();

    v2f xa[16];
#pragma unroll
    for (int s = 0; s < 16; ++s) xa[s] = *(const v2f*)&xw[m * XSTR + 4 * s + 2 * hi];

    const float mu0 = ip[0], a0 = ip[1];
    float ld = -a0;
    if (lane < 16) zw[lane * XSTR + 63] = (xw[lane * XSTR + 0] - mu0) * expf(-a0);

    for (int l = 0; l < LVAL; ++l) {
        __syncthreads();
        for (int i = tid; i < 64 * 16; i += 128) {
            int j = i >> 4, c4 = i & 15;
            float4 v = {0.0f, 0.0f, 0.0f, 0.0f};
            if (j <= l) v = *(const float4*)&W1g[((size_t)l * 63 + j) * 64 + c4 * 4];
            *(float4*)&W1s[j * WSTR + c4 * 4] = v;
            float4 w = *(const float4*)&W2g[((size_t)l * 64 + j) * 64 + c4 * 4];
            *(float4*)&W2s[j * WSTR + c4 * 4] = w;
        }
        if (tid < 64) {
            W3s[tid * W3STR + 0] = W3g[((size_t)l * 64 + tid) * 2 + 0];
            W3s[tid * W3STR + 1] = W3g[((size_t)l * 64 + tid) * 2 + 1];
            b1s[tid] = b1g[l * 64 + tid];
            b2s[tid] = b2g[l * 64 + tid];
        }
        if (tid < 2) b3s[tid] = b3g[l * 2 + tid];
        __syncthreads();

        v8f acc[4];
#pragma unroll
        for (int t = 0; t < 4; ++t) acc[t] = (v8f){0, 0, 0, 0, 0, 0, 0, 0};
#pragma unroll
        for (int s = 0; s < 16; ++s) {
            v2f a = xa[s];
            int kb = 4 * s + 2 * hi;
#pragma unroll
            for (int t = 0; t < 4; ++t) {
                v2f b; b.x = W1s[kb * WSTR + 16 * t + m]; b.y = W1s[(kb + 1) * WSTR + 16 * t + m];
                acc[t] = __builtin_amdgcn_wmma_f32_16x16x4_f32(false, a, false, b, (short)0, acc[t], false, false);
            }
        }
#pragma unroll
        for (int t = 0; t < 4; ++t) {
            float bias = b1s[16 * t + m];
#pragma unroll
            for (int v = 0; v < 8; ++v) hw[(v + 8 * hi) * XSTR + 16 * t + m] = TANHF(acc[t][v] + bias);
        }
        v2f ha[16];
#pragma unroll
        for (int s = 0; s < 16; ++s) ha[s] = *(const v2f*)&hw[m * XSTR + 4 * s + 2 * hi];

#pragma unroll
        for (int t = 0; t < 4; ++t) acc[t] = (v8f){0, 0, 0, 0, 0, 0, 0, 0};
#pragma unroll
        for (int s = 0; s < 16; ++s) {
            v2f a = ha[s];
            int kb = 4 * s + 2 * hi;
#pragma unroll
            for (int t = 0; t < 4; ++t) {
                v2f b; b.x = W2s[kb * WSTR + 16 * t + m]; b.y = W2s[(kb + 1) * WSTR + 16 * t + m];
                acc[t] = __builtin_amdgcn_wmma_f32_16x16x4_f32(false, a, false, b, (short)0, acc[t], false, false);
            }
        }
#pragma unroll
        for (int t = 0; t < 4; ++t) {
            float bias = b2s[16 * t + m];
#pragma unroll
            for (int v = 0; v < 8; ++v) hw[(v + 8 * hi) * XSTR + 16 * t + m] = TANHF(acc[t][v] + bias);
        }
        v2f ha2[16];
#pragma unroll
        for (int s = 0; s < 16; ++s) ha2[s] = *(const v2f*)&hw[m * XSTR + 4 * s + 2 * hi];

        v8f acc3 = (v8f){0, 0, 0, 0, 0, 0, 0, 0};
#pragma unroll
        for (int s = 0; s < 16; ++s) {
            int kb = 4 * s + 2 * hi;
            v2f b; b.x = W3s[kb * W3STR + m]; b.y = W3s[(kb + 1) * W3STR + m];
            acc3 = __builtin_amdgcn_wmma_f32_16x16x4_f32(false, ha2[s], false, b, (short)0, acc3, false, false);
        }
        if (m == 0) {
#pragma unroll
            for (int v = 0; v < 8; ++v) ow[(v + 8 * hi) * 2 + 0] = acc3[v] + b3s[0];
        }
        if (m == 1) {
#pragma unroll
            for (int v = 0; v < 8; ++v) ow[(v + 8 * hi) * 2 + 1] = acc3[v] + b3s[1];
        }
        __syncthreads();
        if (lane < 16) {
            float mu = ow[lane * 2 + 0];
            float al = ow[lane * 2 + 1];
            float zv = (xw[lane * XSTR + (l + 1)] - mu) * expf(-al);
            zw[lane * XSTR + (62 - l)] = zv;
            ld -= al;
        }
    }

    for (int i = lane; i < 16 * 16; i += 32) {
        int r = i >> 4, c4 = i & 15;
        float4 v = *(const float4*)&zw[r * XSTR + c4 * 4];
        *(float4*)&outg[(size_t)(rw + r) * DIMV + c4 * 4] = v;
    }
    if (lane < 16) outg[(size_t)BATCHV * DIMV + rw + lane] = ld;
}

extern "C" void kernel_launch(void* const* d_in, const int* in_sizes, int n_in,
                              void* d_out, int out_size, void* d_ws, size_t ws_size,
                              hipStream_t stream) {
    (void)in_sizes; (void)n_in; (void)out_size;
    const float* xg  = (const float*)d_in[0];
    const float* ip  = (const float*)d_in[1];
    const float* W1g = (const float*)d_in[2];
    const float* b1g = (const float*)d_in[3];
    const float* W2g = (const float*)d_in[4];
    const float* b2g = (const float*)d_in[5];
    const float* W3g = (const float*)d_in[6];
    const float* b3g = (const float*)d_in[7];
    float* outg = (float*)d_out;

    const size_t need = (size_t)(W1T_ELEMS + W2T_ELEMS + W3T_ELEMS) * sizeof(float);
    if (ws_size >= need) {
        float* w1t = (float*)d_ws;
        float* w2t = w1t + W1T_ELEMS;
        float* w3t = w2t + W2T_ELEMS;
        maf_prep_kernel<<<(W1T_ELEMS + 255) / 256, 256, 0, stream>>>(
            W1g, W2g, W3g, w1t, w2t, w3t);
        maf_wmma_tdm_kernel<<<BATCHV / 64, 128, 0, stream>>>(
            xg, ip, w1t, b1g, w2t, b2g, w3t, b3g, outg);
    } else {
        maf_wmma_fallback_kernel<<<BATCHV / 64, 128, 0, stream>>>(
            xg, ip, W1g, b1g, W2g, b2g, W3g, b3g, outg);
    }
}